// GNNLayer_7868380087091
// MI455X (gfx1250) — compile-verified
//
#include <hip/hip_runtime.h>

#define N_NODES 50000
#define TOPK 32
#define IN_C 128
#define OUT_C 64
#define NEG_SLOPE 0.2f
#define BN_EPS 1e-5f
#define RED_BLOCKS 256

typedef float v2f __attribute__((ext_vector_type(2)));
typedef float v8f __attribute__((ext_vector_type(8)));

// ---------------------------------------------------------------------------
// Kernel 1: x_lin[N,64] = x[N,128] @ lin_w[64,128]^T using V_WMMA_F32_16X16X4_F32
// One wave32 per 16-row M-strip; 4 N-tiles of 16 cols; K loop of 32 x 4.
// A 16x4 f32 layout: lanes 0-15 hold K={0,1}, lanes 16-31 K={2,3} (VGPR=K lsb).
// B 4x16 f32 layout: col N = lane%16, K = vgpr + 2*(lane>=16).
// C/D 16x16 f32: VGPR r -> row r + 8*(lane>=16), col lane%16.
// ---------------------------------------------------------------------------
__global__ __launch_bounds__(256) void gemm_xlin(const float* __restrict__ x,
                                                 const float* __restrict__ w,
                                                 float* __restrict__ xlin) {
    const int lane  = threadIdx.x & 31;
    const int mtile = blockIdx.x * (blockDim.x >> 5) + (threadIdx.x >> 5);
    if (mtile >= N_NODES / 16) return;          // uniform per wave -> EXEC stays all-1s
    const int half = lane >> 4;                 // 0: lanes 0-15, 1: lanes 16-31
    const int l16  = lane & 15;

    v8f acc0 = {}; v8f acc1 = {}; v8f acc2 = {}; v8f acc3 = {};

    const float* ar = x + (mtile * 16 + l16) * IN_C + 2 * half;
    const float* b0 = w + (l16 +  0) * IN_C + 2 * half;
    const float* b1 = w + (l16 + 16) * IN_C + 2 * half;
    const float* b2 = w + (l16 + 32) * IN_C + 2 * half;
    const float* b3 = w + (l16 + 48) * IN_C + 2 * half;

#pragma unroll 4
    for (int kc = 0; kc < IN_C / 4; ++kc) {
        v2f a   = *(const v2f*)(ar + 4 * kc);
        v2f vb0 = *(const v2f*)(b0 + 4 * kc);
        v2f vb1 = *(const v2f*)(b1 + 4 * kc);
        v2f vb2 = *(const v2f*)(b2 + 4 * kc);
        v2f vb3 = *(const v2f*)(b3 + 4 * kc);
        acc0 = __builtin_amdgcn_wmma_f32_16x16x4_f32(false, a, false, vb0, (short)0, acc0, false, false);
        acc1 = __builtin_amdgcn_wmma_f32_16x16x4_f32(false, a, false, vb1, (short)0, acc1, false, false);
        acc2 = __builtin_amdgcn_wmma_f32_16x16x4_f32(false, a, false, vb2, (short)0, acc2, false, false);
        acc3 = __builtin_amdgcn_wmma_f32_16x16x4_f32(false, a, false, vb3, (short)0, acc3, false, false);
    }

#pragma unroll
    for (int r = 0; r < 8; ++r) {
        float* o = xlin + (mtile * 16 + r + 8 * half) * OUT_C + l16;
        o[0]  = acc0[r];
        o[16] = acc1[r];
        o[32] = acc2[r];
        o[48] = acc3[r];
    }
}

// ---------------------------------------------------------------------------
// Kernel 2: per-node attention scalars
//   a_i[n] = <x_lin[n],att_i> + <emb[n],att_em_i>
//   a_j[n] = <x_lin[n],att_j> + <emb[n],att_em_j>
// One wave32 per node; lane handles 2 channels; shfl_xor tree reduction.
// ---------------------------------------------------------------------------
__global__ __launch_bounds__(256) void node_scalars(const float* __restrict__ xlin,
                                                    const float* __restrict__ emb,
                                                    const float* __restrict__ att_i,
                                                    const float* __restrict__ att_j,
                                                    const float* __restrict__ att_em_i,
                                                    const float* __restrict__ att_em_j,
                                                    float* __restrict__ a_i,
                                                    float* __restrict__ a_j) {
    const int lane = threadIdx.x & 31;
    const int node = blockIdx.x * (blockDim.x >> 5) + (threadIdx.x >> 5);
    if (node >= N_NODES) return;
    const int c = 2 * lane;
    v2f xl  = *(const v2f*)(xlin + node * OUT_C + c);
    v2f eb  = *(const v2f*)(emb  + node * OUT_C + c);
    v2f ai  = *(const v2f*)(att_i + c);
    v2f aj  = *(const v2f*)(att_j + c);
    v2f aei = *(const v2f*)(att_em_i + c);
    v2f aej = *(const v2f*)(att_em_j + c);
    float si = xl.x * ai.x + xl.y * ai.y + eb.x * aei.x + eb.y * aei.y;
    float sj = xl.x * aj.x + xl.y * aj.y + eb.x * aej.x + eb.y * aej.y;
    for (int off = 16; off; off >>= 1) {
        si += __shfl_xor(si, off);
        sj += __shfl_xor(sj, off);
    }
    if (lane == 0) { a_i[node] = si; a_j[node] = sj; }
}

// ---------------------------------------------------------------------------
// Kernel 3: segment softmax + weighted aggregation. One wave32 per target node.
// Edges of node t are src[t*32 .. t*32+31] plus the self-loop (regular graph).
// Phase 1: lane e -> logit of edge e; leaky-relu; wave max/sum via shfl_xor.
// Phase 2: lane -> 2 channels; loop edges broadcasting (src, w) via shfl.
// x_lin (12.8 MB) is L2-resident, so the random 256B/edge gathers hit L2.
// ---------------------------------------------------------------------------
__global__ __launch_bounds__(256) void attn_agg(const float* __restrict__ xlin,
                                                const int* __restrict__ src,
                                                const float* __restrict__ a_i,
                                                const float* __restrict__ a_j,
                                                const float* __restrict__ bias,
                                                float* __restrict__ out) {
    const int lane = threadIdx.x & 31;
    const int t = blockIdx.x * (blockDim.x >> 5) + (threadIdx.x >> 5);
    if (t >= N_NODES) return;

    const float ai = a_i[t];
    const int  s  = src[t * TOPK + lane];
    float al = ai + a_j[s];
    al = (al >= 0.f) ? al : NEG_SLOPE * al;
    float aself = ai + a_j[t];
    aself = (aself >= 0.f) ? aself : NEG_SLOPE * aself;

    float m = al;
    for (int off = 16; off; off >>= 1) m = fmaxf(m, __shfl_xor(m, off));
    m = fmaxf(m, aself);

    const float w     = __expf(al - m);
    const float wself = __expf(aself - m);
    float ssum = w;
    for (int off = 16; off; off >>= 1) ssum += __shfl_xor(ssum, off);
    ssum += wself;
    const float inv = 1.0f / (ssum + 1e-16f);

    const int c = 2 * lane;
    v2f xs = *(const v2f*)(xlin + t * OUT_C + c);
    float accx = wself * xs.x;
    float accy = wself * xs.y;
#pragma unroll 4
    for (int k = 0; k < TOPK; ++k) {
        const int   sk = __shfl(s, k);
        const float wk = __shfl(w, k);
        v2f xv = *(const v2f*)(xlin + sk * OUT_C + c);
        accx += wk * xv.x;
        accy += wk * xv.y;
    }
    v2f bv = *(const v2f*)(bias + c);
    out[t * OUT_C + c]     = accx * inv + bv.x;
    out[t * OUT_C + c + 1] = accy * inv + bv.y;
}

// ---------------------------------------------------------------------------
// Kernel 4a: per-block partial sums / sums-of-squares per channel (c = i % 64).
// ---------------------------------------------------------------------------
__global__ __launch_bounds__(256) void bn_partial(const float* __restrict__ out,
                                                  float* __restrict__ psum,
                                                  float* __restrict__ psq) {
    __shared__ float ls[256];
    __shared__ float lq[256];
    const int t = threadIdx.x;
    float s = 0.f, q = 0.f;
    const int total = N_NODES * OUT_C;
    for (int i = blockIdx.x * 256 + t; i < total; i += RED_BLOCKS * 256) {
        float v = out[i];
        s += v;
        q += v * v;
    }
    ls[t] = s; lq[t] = q;
    __syncthreads();
    if (t < 64) {
        s = ls[t] + ls[t + 64] + ls[t + 128] + ls[t + 192];
        q = lq[t] + lq[t + 64] + lq[t + 128] + lq[t + 192];
        psum[blockIdx.x * 64 + t] = s;
        psq [blockIdx.x * 64 + t] = q;
    }
}

// ---------------------------------------------------------------------------
// Kernel 4b: finalize -> per-channel scale = gamma*rsqrt(var+eps), shift.
// ---------------------------------------------------------------------------
__global__ void bn_finalize(const float* __restrict__ psum,
                            const float* __restrict__ psq,
                            const float* __restrict__ gamma,
                            const float* __restrict__ beta,
                            float* __restrict__ stats) {
    const int c = threadIdx.x;
    if (c >= OUT_C) return;
    float s = 0.f, q = 0.f;
    for (int b = 0; b < RED_BLOCKS; ++b) {
        s += psum[b * 64 + c];
        q += psq [b * 64 + c];
    }
    const float invn  = 1.0f / (float)N_NODES;
    const float mean  = s * invn;
    const float var   = q * invn - mean * mean;
    const float scale = gamma[c] * rsqrtf(var + BN_EPS);
    stats[c]          = scale;
    stats[64 + c]     = beta[c] - mean * scale;
}

// ---------------------------------------------------------------------------
// Kernel 5: in-place normalize + ReLU.
// ---------------------------------------------------------------------------
__global__ __launch_bounds__(256) void bn_apply(float* __restrict__ out,
                                                const float* __restrict__ stats) {
    const int i = blockIdx.x * blockDim.x + threadIdx.x;
    if (i >= N_NODES * OUT_C) return;
    const int c = i & 63;
    const float v = out[i] * stats[c] + stats[64 + c];
    out[i] = fmaxf(v, 0.f);
}

extern "C" void kernel_launch(void* const* d_in, const int* in_sizes, int n_in,
                              void* d_out, int out_size, void* d_ws, size_t ws_size,
                              hipStream_t stream) {
    (void)in_sizes; (void)n_in; (void)out_size; (void)ws_size;
    const float* x        = (const float*)d_in[0];
    const float* emb      = (const float*)d_in[1];
    const int*   edge     = (const int*)d_in[2];   // [2, N*TOPK]; row 0 = src
    const float* lin_w    = (const float*)d_in[3];
    const float* att_i    = (const float*)d_in[4];
    const float* att_j    = (const float*)d_in[5];
    const float* att_em_i = (const float*)d_in[6];
    const float* att_em_j = (const float*)d_in[7];
    const float* bias     = (const float*)d_in[8];
    const float* gamma    = (const float*)d_in[9];
    const float* beta     = (const float*)d_in[10];
    float* out = (float*)d_out;

    // workspace layout (floats): ~13.3 MB total
    float* ws    = (float*)d_ws;
    float* xlin  = ws;                         // N*64
    float* a_i   = xlin + N_NODES * OUT_C;     // N
    float* a_j   = a_i + N_NODES;              // N
    float* psum  = a_j + N_NODES;              // RED_BLOCKS*64
    float* psq   = psum + RED_BLOCKS * 64;     // RED_BLOCKS*64
    float* stats = psq + RED_BLOCKS * 64;      // 128

    const int waves_per_block = 8;             // 256 threads
    gemm_xlin<<<(N_NODES / 16 + waves_per_block - 1) / waves_per_block, 256, 0, stream>>>(
        x, lin_w, xlin);
    node_scalars<<<(N_NODES + waves_per_block - 1) / waves_per_block, 256, 0, stream>>>(
        xlin, emb, att_i, att_j, att_em_i, att_em_j, a_i, a_j);
    attn_agg<<<(N_NODES + waves_per_block - 1) / waves_per_block, 256, 0, stream>>>(
        xlin, edge, a_i, a_j, bias, out);
    bn_partial<<<RED_BLOCKS, 256, 0, stream>>>(out, psum, psq);
    bn_finalize<<<1, 64, 0, stream>>>(psum, psq, gamma, beta, stats);
    bn_apply<<<(N_NODES * OUT_C + 255) / 256, 256, 0, stream>>>(out, stats);
}